// AttentionPooling_39238821216442
// MI455X (gfx1250) — compile-verified
//
#include <hip/hip_runtime.h>
#include <hip/hip_bf16.h>

typedef float v2f __attribute__((ext_vector_type(2)));
typedef float v8f __attribute__((ext_vector_type(8)));

#define HIDDEN 256
#define HALF   128
#define NT       8        // 8 N-tiles of 16 -> N = 128
#define PSTRIDE 288       // floats per k-pair row: 288 % 64 == 32 -> halves hit disjoint banks
#define CHUNK   256       // nodes per pooling block

// ---- monotonic float<->uint mapping for atomicMax on floats ----
__device__ __forceinline__ unsigned f2ord(float f) {
    int i = __float_as_int(f);
    return (i >= 0) ? ((unsigned)i | 0x80000000u) : ~(unsigned)i;
}
__device__ __forceinline__ float ord2f(unsigned u) {
    int i = (u & 0x80000000u) ? (int)(u & 0x7FFFFFFFu) : ~(int)u;
    return __int_as_float(i);
}

// ---- zero output + segment stats (runs every launch: graph-replay safe) ----
__global__ void init_kernel(float* __restrict__ out, int out_n,
                            unsigned* __restrict__ segmax,
                            float* __restrict__ segsum, int ngr) {
    int i = blockIdx.x * blockDim.x + threadIdx.x;
    if (i < out_n) out[i] = 0.0f;
    if (i < ngr) { segmax[i] = 0u; segsum[i] = 0.0f; }
}

// ---- scores = tanh(x@W1 + b1)@W2 + b2 ; fused segment-max atomic ----
// LDS holds W1 in k-pair interleaved layout:
//   w1p[p*PSTRIDE + 2*col + 0] = W1[2p  ][col]
//   w1p[p*PSTRIDE + 2*col + 1] = W1[2p+1][col]
// so every WMMA B fragment is ONE contiguous ds_load_b64 into an aligned VGPR pair.
__global__ __launch_bounds__(256) void score_kernel(
    const float* __restrict__ x, const int* __restrict__ batch,
    const float* __restrict__ W1, const float* __restrict__ b1,
    const float* __restrict__ W2, const float* __restrict__ b2,
    float* __restrict__ s, unsigned* __restrict__ segmax, int n_nodes) {

    extern __shared__ float w1p[];   // (HIDDEN/2) x PSTRIDE floats = 144 KB

    // cooperative load: float4 global reads, scattered pair-layout LDS stores
    {
        const float4* src = (const float4*)W1;     // 256x128 row-major -> 8192 float4
        for (int idx = threadIdx.x; idx < (HIDDEN * HALF) / 4; idx += blockDim.x) {
            float4 v = src[idx];
            int k    = idx >> 5;                    // row (128 floats = 32 float4 per row)
            int col  = (idx & 31) * 4;              // starting column
            float* d = w1p + (k >> 1) * PSTRIDE + (k & 1);
            d[2 * (col + 0)] = v.x;
            d[2 * (col + 1)] = v.y;
            d[2 * (col + 2)] = v.z;
            d[2 * (col + 3)] = v.w;
        }
    }
    __syncthreads();

    const int wave = threadIdx.x >> 5;
    const int lane = threadIdx.x & 31;
    const int tile = blockIdx.x * 8 + wave;
    if (tile * 16 >= n_nodes) return;

    const int row0  = tile * 16;
    const int m     = lane & 15;            // M (A) / N-sub (B) index
    const int hi    = lane >> 4;            // 0 for lanes 0-15, 1 for lanes 16-31
    const int khalf = hi << 1;              // K offset within the 4-wide step

    const float* xrow = x + (size_t)(row0 + m) * HIDDEN + khalf;
    // B base: pair row (k0/2 + hi), column pair offset 2*m
    const float* bp = w1p + (size_t)hi * PSTRIDE + 2 * m;

    v8f acc[NT] = {};

    for (int k0 = 0; k0 < HIDDEN; k0 += 4) {
        v2f a = *(const v2f*)(xrow + k0);                    // K = k0+khalf, k0+khalf+1
        const float* bk = bp + (k0 >> 1) * PSTRIDE;
#pragma unroll
        for (int n = 0; n < NT; ++n) {
            v2f b = *(const v2f*)(bk + n * 32);              // {W1[k][16n+m], W1[k+1][16n+m]}
            acc[n] = __builtin_amdgcn_wmma_f32_16x16x4_f32(
                false, a, false, b, (short)0, acc[n], false, false);
        }
    }

    // epilogue: per-lane column = 16n + m ; row = r + (hi ? 8 : 0)
    float b1f[NT], w2f[NT];
#pragma unroll
    for (int n = 0; n < NT; ++n) {
        b1f[n] = b1[n * 16 + m];
        w2f[n] = W2[n * 16 + m];
    }
    const float b2v = b2[0];

#pragma unroll
    for (int r = 0; r < 8; ++r) {
        float p = 0.0f;
#pragma unroll
        for (int n = 0; n < NT; ++n)
            p += tanhf(acc[n][r] + b1f[n]) * w2f[n];
        // reduce across the 16 lanes of this half
        p += __shfl_xor(p, 1, 16);
        p += __shfl_xor(p, 2, 16);
        p += __shfl_xor(p, 4, 16);
        p += __shfl_xor(p, 8, 16);
        float score = p + b2v;
        int row = row0 + r + (hi << 3);
        if (m == 0) {                       // lanes 0 and 16 write rows r and r+8
            s[row] = score;
            atomicMax(&segmax[batch[row]], f2ord(score));
        }
    }
}

// ---- e = exp(s - max[g]) in place; segment sums (sorted-batch fast path) ----
__global__ void expsum_kernel(float* __restrict__ e, const int* __restrict__ batch,
                              const unsigned* __restrict__ segmax,
                              float* __restrict__ segsum, int n) {
    int i = blockIdx.x * blockDim.x + threadIdx.x;
    if (i >= n) return;                     // n % 32 == 0: never splits a wave
    int g = batch[i];
    float mx = ord2f(segmax[g]);
    if (!isfinite(mx)) mx = 0.0f;           // matches jnp.where(isfinite, ..., 0)
    float ev = expf(e[i] - mx);
    e[i] = ev;

    int lane = threadIdx.x & 31;
    int g0 = __shfl(g, 0);
    int g1 = __shfl(g, 31);
    if (g0 == g1) {                         // whole wave in one segment (common: sorted)
        float tot = ev;
        tot += __shfl_xor(tot, 16);
        tot += __shfl_xor(tot, 8);
        tot += __shfl_xor(tot, 4);
        tot += __shfl_xor(tot, 2);
        tot += __shfl_xor(tot, 1);
        if (lane == 0) atomicAdd(&segsum[g], tot);
    } else {
        atomicAdd(&segsum[g], ev);
    }
}

// ---- w = e / sum[g] in place ----
__global__ void wnorm_kernel(float* __restrict__ e, const int* __restrict__ batch,
                             const float* __restrict__ segsum, int n) {
    int i = blockIdx.x * blockDim.x + threadIdx.x;
    if (i < n) e[i] = e[i] / segsum[batch[i]];
}

// ---- out[g,:] += w_i * x_i  (thread = column; boundary-only atomics) ----
__global__ __launch_bounds__(256) void pool_kernel(
    const float* __restrict__ x, const int* __restrict__ batch,
    const float* __restrict__ w, float* __restrict__ out, int n) {
    int c = threadIdx.x;                    // 0..255 column
    int start = blockIdx.x * CHUNK;
    if (start >= n) return;
    int end = min(start + CHUNK, n);

    int cur = batch[start];                 // uniform -> scalar load
    float acc = 0.0f;
    for (int i = start; i < end; ++i) {
        int g = batch[i];
        if (g != cur) {
            atomicAdd(&out[(size_t)cur * HIDDEN + c], acc);
            acc = 0.0f;
            cur = g;
        }
        acc += x[(size_t)i * HIDDEN + c] * w[i];
    }
    atomicAdd(&out[(size_t)cur * HIDDEN + c], acc);
}

extern "C" void kernel_launch(void* const* d_in, const int* in_sizes, int n_in,
                              void* d_out, int out_size, void* d_ws, size_t ws_size,
                              hipStream_t stream) {
    const float* x   = (const float*)d_in[0];
    const int* batch = (const int*)d_in[1];
    const float* W1  = (const float*)d_in[2];
    const float* b1  = (const float*)d_in[3];
    const float* W2  = (const float*)d_in[4];
    const float* b2  = (const float*)d_in[5];
    float* out = (float*)d_out;

    const int n   = in_sizes[1];            // N_NODES
    const int ngr = out_size / HIDDEN;      // N_GRAPHS

    // workspace layout: scores/e/w (n floats), segmax (u32), segsum (f32)
    float*    s      = (float*)d_ws;
    unsigned* segmax = (unsigned*)(s + ((n + 63) & ~63));
    float*    segsum = (float*)(segmax + ((ngr + 63) & ~63));

    int initN = out_size > ngr ? out_size : ngr;
    init_kernel<<<(initN + 255) / 256, 256, 0, stream>>>(out, out_size, segmax, segsum, ngr);

    int ntiles  = (n + 15) / 16;
    int nblocks = (ntiles + 7) / 8;
    size_t lds = (size_t)(HIDDEN / 2) * PSTRIDE * sizeof(float);   // 144 KB
    score_kernel<<<nblocks, 256, lds, stream>>>(x, batch, W1, b1, W2, b2, s, segmax, n);

    expsum_kernel<<<(n + 255) / 256, 256, 0, stream>>>(s, batch, segmax, segsum, n);
    wnorm_kernel<<<(n + 255) / 256, 256, 0, stream>>>(s, batch, segsum, n);
    pool_kernel<<<(n + CHUNK - 1) / CHUNK, 256, 0, stream>>>(x, batch, s, out, n);
}